// MMA_16372415332361
// MI455X (gfx1250) — compile-verified
//
#include <hip/hip_runtime.h>
#include <math.h>

#define NN 50000
#define NE 800000
#define FD 128
#define EA (NE + NN)   // 850000 total edges incl. self loops

typedef __attribute__((ext_vector_type(2))) float v2f;
typedef __attribute__((ext_vector_type(8))) float v8f;

__device__ __forceinline__ float sigmoidf(float t) {
    return 1.0f / (1.0f + __expf(-t));
}

__device__ __forceinline__ void atomic_add_f32(float* p, float v) {
    __hip_atomic_fetch_add(p, v, __ATOMIC_RELAXED, __HIP_MEMORY_SCOPE_AGENT);
}

// ---------------------------------------------------------------------------
// 1) init: x_new = x, update = 0, degrees = 0
// ---------------------------------------------------------------------------
__global__ void init_ws(const float* __restrict__ x,
                        float* __restrict__ deg_in, float* __restrict__ deg_out,
                        float* __restrict__ x_new, float* __restrict__ update) {
    int i = blockIdx.x * blockDim.x + threadIdx.x;
    if (i < NN * FD) {
        x_new[i]  = x[i];
        update[i] = 0.0f;
    }
    if (i < NN) { deg_in[i] = 0.0f; deg_out[i] = 0.0f; }
}

// ---------------------------------------------------------------------------
// 2) degree accumulation (segment sums over edges)
// ---------------------------------------------------------------------------
__global__ void degrees_kernel(const int* __restrict__ ei, const float* __restrict__ ew,
                               float* __restrict__ deg_in, float* __restrict__ deg_out) {
    int e = blockIdx.x * blockDim.x + threadIdx.x;
    if (e < NE) {
        int s = ei[2 * e];
        int d = ei[2 * e + 1];
        float w = ew[e];
        atomic_add_f32(&deg_in[s], w);
        atomic_add_f32(&deg_out[d], w);
    }
}

// ---------------------------------------------------------------------------
// 3) xa = x @ wa, xb = x @ wb via V_WMMA_F32_16X16X4_F32
//    One wave per 16x16 output tile; 8 waves/block cover the 8 col-tiles of F=128.
//    grid.x = NN/16 = 3125 row tiles (exact -> EXEC always all-ones).
// ---------------------------------------------------------------------------
__global__ void gemm_gates(const float* __restrict__ x, const float* __restrict__ wm0,
                           float* __restrict__ xa, float* __restrict__ xb) {
    const int lane  = threadIdx.x & 31;
    const int wv    = threadIdx.x >> 5;     // 0..7 : column tile
    const int m0    = blockIdx.x * 16;      // row tile base
    const int c0    = wv * 16;
    const int ln15  = lane & 15;
    const int khalf = (lane >> 4) * 2;      // lanes 0-15: K=+0,+1 ; lanes 16-31: K=+2,+3

    const int row = m0 + ln15;              // A: M = lane&15
    const int col = c0 + ln15;              // B/D: N = lane&15

    v8f accA = {0.f, 0.f, 0.f, 0.f, 0.f, 0.f, 0.f, 0.f};
    v8f accB = accA;

    for (int k0 = 0; k0 < FD; k0 += 4) {
        const int kk = k0 + khalf;
        v2f a;
        a.x = x[row * FD + kk];
        a.y = x[row * FD + kk + 1];
        v2f bA, bB;
        bA.x = wm0[kk * FD + col];              // wa = weights_mask0[0:F]
        bA.y = wm0[(kk + 1) * FD + col];
        bB.x = wm0[(FD + kk) * FD + col];       // wb = weights_mask0[F:2F]
        bB.y = wm0[(FD + kk + 1) * FD + col];
        accA = __builtin_amdgcn_wmma_f32_16x16x4_f32(false, a, false, bA, (short)0,
                                                     accA, false, false);
        accB = __builtin_amdgcn_wmma_f32_16x16x4_f32(false, a, false, bB, (short)0,
                                                     accB, false, false);
    }

    const int rb = m0 + (lane >> 4) * 8;    // lanes 0-15: rows 0-7 ; 16-31: rows 8-15
#pragma unroll
    for (int r = 0; r < 8; ++r) {
        xa[(rb + r) * FD + col] = accA[r];
        xb[(rb + r) * FD + col] = accB[r];
    }
}

// ---------------------------------------------------------------------------
// 4) mask = sigmoid(xa[s]+xb[d]); x_new[s] += mask * x[d]
//    One wave per edge, each lane covers 4 features (float4 gathers, f32 atomics).
// ---------------------------------------------------------------------------
__global__ void edge_mask_scatter(const int* __restrict__ ei,
                                  const float* __restrict__ x,
                                  const float* __restrict__ xa,
                                  const float* __restrict__ xb,
                                  float* __restrict__ x_new) {
    const int lane = threadIdx.x & 31;
    const int e = blockIdx.x * 8 + (threadIdx.x >> 5);
    if (e >= EA) return;
    int s, d;
    if (e < NE) { s = ei[2 * e]; d = ei[2 * e + 1]; }
    else        { s = e - NE;    d = s; }

    const int f0 = lane * 4;
    const float4 va = *(const float4*)(xa + (size_t)s * FD + f0);
    const float4 vb = *(const float4*)(xb + (size_t)d * FD + f0);
    const float4 vx = *(const float4*)(x  + (size_t)d * FD + f0);
    float* dst = x_new + (size_t)s * FD + f0;
    atomic_add_f32(dst + 0, sigmoidf(va.x + vb.x) * vx.x);
    atomic_add_f32(dst + 1, sigmoidf(va.y + vb.y) * vx.y);
    atomic_add_f32(dst + 2, sigmoidf(va.z + vb.z) * vx.z);
    atomic_add_f32(dst + 3, sigmoidf(va.w + vb.w) * vx.w);
}

// ---------------------------------------------------------------------------
// 5) update[d] += ew_norm * x_new[s],  ew_norm = w / sqrt((deg_in[s]+1)*(deg_out[d]+1))
// ---------------------------------------------------------------------------
__global__ void edge_aggregate(const int* __restrict__ ei, const float* __restrict__ ew,
                               const float* __restrict__ deg_in,
                               const float* __restrict__ deg_out,
                               const float* __restrict__ x_new,
                               float* __restrict__ update) {
    const int lane = threadIdx.x & 31;
    const int e = blockIdx.x * 8 + (threadIdx.x >> 5);
    if (e >= EA) return;
    int s, d; float w;
    if (e < NE) { s = ei[2 * e]; d = ei[2 * e + 1]; w = ew[e]; }
    else        { s = e - NE;    d = s;             w = 1.0f;  }

    const float ewn = w * rsqrtf((deg_in[s] + 1.0f) * (deg_out[d] + 1.0f));

    const int f0 = lane * 4;
    const float4 vn = *(const float4*)(x_new + (size_t)s * FD + f0);
    float* dst = update + (size_t)d * FD + f0;
    atomic_add_f32(dst + 0, ewn * vn.x);
    atomic_add_f32(dst + 1, ewn * vn.y);
    atomic_add_f32(dst + 2, ewn * vn.z);
    atomic_add_f32(dst + 3, ewn * vn.w);
}

// ---------------------------------------------------------------------------
// 6) out = relu(update @ w_lin^T + b_lin) via V_WMMA_F32_16X16X4_F32
//    B[k][n] = w_lin[n][k]  (row of w_lin is contiguous -> contiguous B loads)
// ---------------------------------------------------------------------------
__global__ void gemm_out(const float* __restrict__ upd, const float* __restrict__ wlin,
                         const float* __restrict__ blin, float* __restrict__ out) {
    const int lane  = threadIdx.x & 31;
    const int wv    = threadIdx.x >> 5;
    const int m0    = blockIdx.x * 16;
    const int c0    = wv * 16;
    const int ln15  = lane & 15;
    const int khalf = (lane >> 4) * 2;

    const int row = m0 + ln15;
    const int col = c0 + ln15;

    v8f acc = {0.f, 0.f, 0.f, 0.f, 0.f, 0.f, 0.f, 0.f};

    for (int k0 = 0; k0 < FD; k0 += 4) {
        const int kk = k0 + khalf;
        v2f a;
        a.x = upd[row * FD + kk];
        a.y = upd[row * FD + kk + 1];
        v2f b;
        b.x = wlin[col * FD + kk];      // B[k][n] = w_lin[n][k]
        b.y = wlin[col * FD + kk + 1];
        acc = __builtin_amdgcn_wmma_f32_16x16x4_f32(false, a, false, b, (short)0,
                                                    acc, false, false);
    }

    const float bias = blin[col];
    const int rb = m0 + (lane >> 4) * 8;
#pragma unroll
    for (int r = 0; r < 8; ++r) {
        out[(rb + r) * FD + col] = fmaxf(acc[r] + bias, 0.0f);
    }
}

// ---------------------------------------------------------------------------
extern "C" void kernel_launch(void* const* d_in, const int* in_sizes, int n_in,
                              void* d_out, int out_size, void* d_ws, size_t ws_size,
                              hipStream_t stream) {
    const float* x    = (const float*)d_in[0];   // (N,F)
    const float* ew   = (const float*)d_in[1];   // (E,)
    const float* wm0  = (const float*)d_in[2];   // (2F,F)
    const float* wlin = (const float*)d_in[3];   // (F,F)
    const float* blin = (const float*)d_in[4];   // (F,)
    const int*   ei   = (const int*)d_in[5];     // (E,2)
    float* out = (float*)d_out;

    float* ws      = (float*)d_ws;
    float* deg_in  = ws;
    float* deg_out = ws + NN;
    float* xa      = ws + 2 * NN;
    float* xb      = xa + (size_t)NN * FD;
    float* x_new   = xb + (size_t)NN * FD;
    float* update  = x_new + (size_t)NN * FD;

    init_ws<<<(NN * FD + 255) / 256, 256, 0, stream>>>(x, deg_in, deg_out, x_new, update);
    degrees_kernel<<<(NE + 255) / 256, 256, 0, stream>>>(ei, ew, deg_in, deg_out);
    gemm_gates<<<NN / 16, 256, 0, stream>>>(x, wm0, xa, xb);
    edge_mask_scatter<<<(EA + 7) / 8, 256, 0, stream>>>(ei, x, xa, xb, x_new);
    edge_aggregate<<<(EA + 7) / 8, 256, 0, stream>>>(ei, ew, deg_in, deg_out, x_new, update);
    gemm_out<<<NN / 16, 256, 0, stream>>>(update, wlin, blin, out);
}